// CubeLens_20581483282930
// MI455X (gfx1250) — compile-verified
//
#include <hip/hip_runtime.h>
#include <stdint.h>

#define C_CH 64
#define NS   256
#define NL   192
#define UPS  4
#define CPB  16           // channels per block (grid.z = 64/16)
#define TX   16           // output tile width per block
#define TY   8            // output tile height per block
#define TPB  (TX*TY)      // 128 threads = 4 wave32

#ifndef __has_builtin
#define __has_builtin(x) 0
#endif

#if defined(__HIP_DEVICE_COMPILE__) && __has_builtin(__builtin_amdgcn_global_load_async_to_lds_b32)
#define USE_ASYNC_LDS 1
#else
#define USE_ASYNC_LDS 0
#endif

#if USE_ASYNC_LDS
typedef __attribute__((address_space(1))) int GINT;   // global (AS1) int
typedef __attribute__((address_space(3))) int LINT;   // LDS    (AS3) int
__device__ __forceinline__ void async_g2l_b32(const void* g, uint32_t lds_off) {
  __builtin_amdgcn_global_load_async_to_lds_b32(
      (GINT*)(uintptr_t)g, (LINT*)(uintptr_t)lds_off, 0, 0);
}
#if __has_builtin(__builtin_amdgcn_s_wait_asynccnt)
#define WAIT_ASYNC(n) __builtin_amdgcn_s_wait_asynccnt(n)
#else
#define WAIT_ASYNC(n) asm volatile("s_wait_asynccnt %0" :: "i"(n) : "memory")
#endif
#endif

__global__ __launch_bounds__(TPB) void cube_lens_kernel(
    const float* __restrict__ src,      // (64, 256, 256) f32
    const float* __restrict__ thetaE,   // scalar
    float* __restrict__ out)            // (64, 192, 192) f32
{
  const int tid = threadIdx.x;
  const int tx  = tid & (TX - 1);
  const int ty  = tid / TX;
  const int J   = blockIdx.x * TX + tx;   // output x
  const int I   = blockIdx.y * TY + ty;   // output y
  const int c0  = blockIdx.z * CPB;
  const float tE = thetaE[0];

  // ---- Geometry for the 16 lensed samples of this output pixel (registers) ----
  int   ofs[16];
  float w00[16], w01[16], w10[16], w11[16];

#pragma unroll
  for (int di = 0; di < 4; ++di) {
#pragma unroll
    for (int dj = 0; dj < 4; ++dj) {
      const int k = di * 4 + dj;
      // coords = (idx - (768-1)/2) * (PS_LENS/UPS)
      const float thx = ((float)(J * UPS + dj) - 383.5f) * 0.01f;
      const float thy = ((float)(I * UPS + di) - 383.5f) * 0.01f;
      const float r   = sqrtf(thx * thx + thy * thy) + 1e-8f;
      const float bx  = thx - (tE * thx) / r;
      const float by  = thy - (tE * thy) / r;
      const float fx  = bx / 0.02f + 127.5f;   // /PS_SRC + (NS-1)/2
      const float fy  = by / 0.02f + 127.5f;
      const bool inb = (fx >= 0.0f) && (fx <= 255.0f) && (fy >= 0.0f) && (fy <= 255.0f);
      int x0 = (int)floorf(fx); x0 = x0 < 0 ? 0 : (x0 > NS - 2 ? NS - 2 : x0);
      int y0 = (int)floorf(fy); y0 = y0 < 0 ? 0 : (y0 > NS - 2 ? NS - 2 : y0);
      float wx = fx - (float)x0; wx = wx < 0.0f ? 0.0f : (wx > 1.0f ? 1.0f : wx);
      float wy = fy - (float)y0; wy = wy < 0.0f ? 0.0f : (wy > 1.0f ? 1.0f : wy);
      const float s  = inb ? 0.0625f : 0.0f;   // fold 1/16 pooling + OOB mask
      const float ox = 1.0f - wx, oy = 1.0f - wy;
      w00[k] = oy * ox * s;  w01[k] = oy * wx * s;
      w10[k] = wy * ox * s;  w11[k] = wy * wx * s;
      ofs[k] = y0 * NS + x0;
    }
  }

#if USE_ASYNC_LDS
  // ---- Async global->LDS double-buffered gather, 8-sample half-batches ----
  __shared__ float stage[2][TPB * 32];           // 32 KB total
  const uint32_t lds0 = (uint32_t)(uintptr_t)&stage[0][tid * 32];
  const uint32_t lds1 = (uint32_t)(uintptr_t)&stage[1][tid * 32];

  auto issue_half = [&](int batch) {
    const int cc   = batch >> 1;
    const int half = batch & 1;
    const uint32_t lb = half ? lds1 : lds0;
    const float* base = src + (size_t)(c0 + cc) * (NS * NS);
#pragma unroll
    for (int k = 0; k < 8; ++k) {
      const int s = half * 8 + k;
      const float* g = base + ofs[s];
      async_g2l_b32(g,          lb + k * 16 + 0);   // v00
      async_g2l_b32(g + 1,      lb + k * 16 + 4);   // v01
      async_g2l_b32(g + NS,     lb + k * 16 + 8);   // v10
      async_g2l_b32(g + NS + 1, lb + k * 16 + 12);  // v11
    }
  };

  issue_half(0);
  float acc = 0.0f;
  const int nb = CPB * 2;
#pragma unroll 2
  for (int batch = 0; batch < nb; ++batch) {
    const int half = batch & 1;
    if (batch + 1 < nb) { issue_half(batch + 1); WAIT_ASYNC(32); }  // oldest 32 done (in-order)
    else                { WAIT_ASYNC(0); }
    asm volatile("" ::: "memory");
    const float* sl = half ? &stage[1][tid * 32] : &stage[0][tid * 32];
#pragma unroll
    for (int k = 0; k < 8; ++k) {
      const int s = half * 8 + k;
      const float4 v = *(const float4*)(sl + k * 4);  // ds_load_b128
      acc = fmaf(w00[s], v.x, acc);
      acc = fmaf(w01[s], v.y, acc);
      acc = fmaf(w10[s], v.z, acc);
      acc = fmaf(w11[s], v.w, acc);
    }
    if (half) {
      const int cc = batch >> 1;
      out[((size_t)(c0 + cc) * NL + I) * NL + J] = acc;
      acc = 0.0f;
    }
  }
#else
  // ---- Fallback: direct gathered loads ----
  for (int cc = 0; cc < CPB; ++cc) {
    const float* base = src + (size_t)(c0 + cc) * (NS * NS);
    float acc = 0.0f;
#pragma unroll
    for (int k = 0; k < 16; ++k) {
      const float* g = base + ofs[k];
      acc = fmaf(w00[k], g[0],      acc);
      acc = fmaf(w01[k], g[1],      acc);
      acc = fmaf(w10[k], g[NS],     acc);
      acc = fmaf(w11[k], g[NS + 1], acc);
    }
    out[((size_t)(c0 + cc) * NL + I) * NL + J] = acc;
  }
#endif
}

extern "C" void kernel_launch(void* const* d_in, const int* in_sizes, int n_in,
                              void* d_out, int out_size, void* d_ws, size_t ws_size,
                              hipStream_t stream) {
  (void)in_sizes; (void)n_in; (void)out_size; (void)d_ws; (void)ws_size;
  const float* src = (const float*)d_in[0];
  const float* tE  = (const float*)d_in[1];
  float* out = (float*)d_out;
  dim3 grid(NL / TX, NL / TY, C_CH / CPB);   // 12 x 24 x 4
  dim3 block(TPB);                            // 128 threads
  hipLaunchKernelGGL(cube_lens_kernel, grid, block, 0, stream, src, tE, out);
}